// TotalLoss_15_3959959847226
// MI455X (gfx1250) — compile-verified
//
#include <hip/hip_runtime.h>
#include <hip/hip_bf16.h>
#include <math.h>

typedef __attribute__((ext_vector_type(2))) float v2f;
typedef __attribute__((ext_vector_type(8))) float v8f;

#define B_    64
#define NC_   20
#define K_    150
#define C_    2048
#define TLEN_ 750
#define EPSF  1e-7f
#define TNCE  0.07f
#define ETHETA 0.1f
#define THETA_ 0.1f
#define EALPHA 1.0f

// ---------------- K0: zero scalar accumulators ----------------
__global__ void k0_zero(float* acc) {
    if (threadIdx.x < 8) acc[threadIdx.x] = 0.0f;
}

// ---------------- K1: means over K for HA/EA/HB/EB ----------------
// grid (C/256, B), block 256. Coalesced streaming of all four tensors.
__global__ __launch_bounds__(256) void k1_means(
    const float* __restrict__ HA, const float* __restrict__ EA,
    const float* __restrict__ HB, const float* __restrict__ EB,
    float* __restrict__ qA, float* __restrict__ kA,
    float* __restrict__ qB, float* __restrict__ kB)
{
    int c = blockIdx.x * 256 + threadIdx.x;
    int n = blockIdx.y;
    size_t base = (size_t)n * K_ * C_ + c;
    float s0 = 0.f, s1 = 0.f, s2 = 0.f, s3 = 0.f;
    for (int k = 0; k < K_; ++k) {
        size_t o = base + (size_t)k * C_;
        s0 += HA[o]; s1 += EA[o]; s2 += HB[o]; s3 += EB[o];
    }
    size_t oo = (size_t)n * C_ + c;
    const float inv = 1.0f / (float)K_;
    qA[oo] = s0 * inv; kA[oo] = s1 * inv; qB[oo] = s2 * inv; kB[oo] = s3 * inv;
}

// ---------------- K2: per-n norms and positive logits ----------------
__global__ __launch_bounds__(256) void k2_pos(
    const float* __restrict__ qA, const float* __restrict__ kA,
    const float* __restrict__ qB, const float* __restrict__ kB,
    float* __restrict__ lposA, float* __restrict__ lposB,
    float* __restrict__ invQA, float* __restrict__ invQB)
{
    int n = blockIdx.x;
    int t = threadIdx.x;
    float s[6] = {0.f, 0.f, 0.f, 0.f, 0.f, 0.f};
    for (int c = t; c < C_; c += 256) {
        size_t o = (size_t)n * C_ + c;
        float a = qA[o], b = kA[o], x = qB[o], y = kB[o];
        s[0] += a * a; s[1] += b * b; s[2] += a * b;
        s[3] += x * x; s[4] += y * y; s[5] += x * y;
    }
    __shared__ float sh[6][256];
    for (int i = 0; i < 6; ++i) sh[i][t] = s[i];
    __syncthreads();
    for (int str = 128; str > 0; str >>= 1) {
        if (t < str)
            for (int i = 0; i < 6; ++i) sh[i][t] += sh[i][t + str];
        __syncthreads();
    }
    if (t == 0) {
        float nq = sqrtf(sh[0][0]), nk = sqrtf(sh[1][0]);
        lposA[n] = sh[2][0] / (nq * nk);
        invQA[n] = 1.0f / nq;
        float nq2 = sqrtf(sh[3][0]), nk2 = sqrtf(sh[4][0]);
        lposB[n] = sh[5][0] / (nq2 * nk2);
        invQB[n] = 1.0f / nq2;
    }
}

// ---------------- K3: WMMA Gram kernel: q·E_k and ||E_k||^2 ----------------
// One wave per (ktile, n, which). Row 0 = q[n], rows 1..15 = 15 E rows.
// A-fragment == B-fragment for a Gram product: lane holds vec[lane&15][c+koff..+1].
// D[0][j] = q·E_j, D[j][j] = ||E_j||^2 after 512 chained v_wmma_f32_16x16x4_f32.
__global__ __launch_bounds__(32) void k3_gram(
    const float* __restrict__ qA, const float* __restrict__ qB,
    const float* __restrict__ EA, const float* __restrict__ EB,
    float* __restrict__ dotA, float* __restrict__ nrmA,
    float* __restrict__ dotB, float* __restrict__ nrmB)
{
    int tt = blockIdx.x;      // 0..9  (15 k's per tile)
    int n  = blockIdx.y;      // 0..63
    int w  = blockIdx.z;      // 0: q=qA vs EB ; 1: q=qB vs EA
    const float* q = w ? qB : qA;
    const float* E = w ? EA : EB;
    float* dOut = w ? dotB : dotA;
    float* nOut = w ? nrmB : nrmA;

    int l = threadIdx.x;          // 0..31
    int m = l & 15;               // row/col index in tile
    int koff = (l >> 4) << 1;     // 0 or 2
    const float* base = (m == 0)
        ? (q + (size_t)n * C_)
        : (E + ((size_t)n * K_ + (size_t)(tt * 15 + m - 1)) * C_);

    v8f c = {};
    for (int t = 0; t < C_ / 4; ++t) {
        v2f a = *(const v2f*)(base + t * 4 + koff);
        // D += A x A^T over this 4-wide K chunk (Gram accumulation)
        c = __builtin_amdgcn_wmma_f32_16x16x4_f32(
                false, a, false, a, (short)0, c, false, false);
    }
    float cd[8];
#pragma unroll
    for (int i = 0; i < 8; ++i) cd[i] = c[i];

    size_t ob = (size_t)n * K_ + (size_t)tt * 15;
    if (l >= 1 && l <= 15) {
        dOut[ob + (l - 1)] = cd[0];             // D[0][l] = q . E_l
        if (l <= 7) nOut[ob + (l - 1)] = cd[l]; // D[l][l], l=1..7
    } else if (l >= 24) {
        int j = l - 16;                         // 8..15
        nOut[ob + (j - 1)] = cd[l - 24];        // D[j][j], j=8..15
    }
}

// ---------------- K4: windowed actioness loss ----------------
__device__ inline float reduce64(float v, float* sh2) {
    for (int off = 16; off > 0; off >>= 1) v += __shfl_down(v, off, 32);
    int lane = threadIdx.x & 31, wid = threadIdx.x >> 5;
    if (lane == 0) sh2[wid] = v;
    __syncthreads();
    float r = (threadIdx.x == 0) ? (sh2[0] + sh2[1]) : 0.0f;
    __syncthreads();
    return r;
}

__global__ __launch_bounds__(64) void k4_act(
    const float* __restrict__ a0, const float* __restrict__ a2, float* acc)
{
    int t = blockIdx.x;    // 0..749
    int b = threadIdx.x;   // 0..63
    __shared__ float sh2[2];
    __shared__ float s1[11], s2[11];

    float x0 = a0[b * TLEN_ + t];
    float x2 = a2[b * TLEN_ + t];

    for (int j = 0; j < 11; ++j) {
        int i = t + j - 6;                 // padded index minus 6
        int si = i < 0 ? 0 : (i > TLEN_ - 1 ? TLEN_ - 1 : i);
        float d1 = x0 - a0[b * TLEN_ + si];
        float d2 = fabsf(x2 - a2[b * TLEN_ + si]);
        float r1 = reduce64(d1 * d1, sh2);
        float r2 = reduce64(d2, sh2);
        if (threadIdx.x == 0) { s1[j] = r1; s2[j] = r2; }
    }
    float d = x0 - x2;
    float r3 = reduce64(d * d, sh2);
    if (threadIdx.x == 0) {
        float ws = 0.0f;
        for (int j = 0; j < 11; ++j)
            ws += expf(-0.5f * s1[j]) * (s2[j] * (1.0f / (float)B_));
        atomicAdd(&acc[3], ws);
        atomicAdd(&acc[4], r3);
    }
}

// ---------------- K5: classification BCE ----------------
__global__ __launch_bounds__(256) void k5_cls(
    const float* __restrict__ vs, const float* __restrict__ lab, float* acc)
{
    float s = 0.0f;
    for (int i = threadIdx.x; i < B_ * NC_; i += 256) {
        int n = i / NC_;
        float rs = 0.0f;
        for (int j = 0; j < NC_; ++j) rs += lab[n * NC_ + j];
        float L = lab[i] / rs;
        float v = fminf(fmaxf(vs[i], EPSF), 1.0f - EPSF);
        s += L * logf(v) + (1.0f - L) * log1pf(-v);
    }
    __shared__ float sh[256];
    sh[threadIdx.x] = s;
    __syncthreads();
    for (int str = 128; str > 0; str >>= 1) {
        if (threadIdx.x < str) sh[threadIdx.x] += sh[threadIdx.x + str];
        __syncthreads();
    }
    if (threadIdx.x == 0) acc[0] = -sh[0] / (float)(B_ * NC_);
}

// ---------------- K6: reliability MSE ----------------
__global__ __launch_bounds__(256) void k6_rel(
    const float* __restrict__ rs, const float* __restrict__ rd, float* acc)
{
    const int N = 2 * B_ * TLEN_;
    float s1 = 0.f, s2 = 0.f;
    for (int i = blockIdx.x * blockDim.x + threadIdx.x; i < N;
         i += gridDim.x * blockDim.x) {
        float a = 1.0f - rs[i]; s1 += a * a;
        float b = rd[i];        s2 += b * b;
    }
    __shared__ float sh[2][256];
    sh[0][threadIdx.x] = s1; sh[1][threadIdx.x] = s2;
    __syncthreads();
    for (int str = 128; str > 0; str >>= 1) {
        if (threadIdx.x < str) {
            sh[0][threadIdx.x] += sh[0][threadIdx.x + str];
            sh[1][threadIdx.x] += sh[1][threadIdx.x + str];
        }
        __syncthreads();
    }
    if (threadIdx.x == 0) {
        atomicAdd(&acc[1], sh[0][0]);
        atomicAdd(&acc[2], sh[1][0]);
    }
}

// ---------------- K7: NCE softmax + final assembly ----------------
__global__ __launch_bounds__(64) void k7_final(
    const float* __restrict__ att,
    const float* __restrict__ lposA, const float* __restrict__ lposB,
    const float* __restrict__ invQA, const float* __restrict__ invQB,
    const float* __restrict__ dotA, const float* __restrict__ nrmA,
    const float* __restrict__ dotB, const float* __restrict__ nrmB,
    const float* __restrict__ acc, float* __restrict__ out)
{
    int n = threadIdx.x;
    __shared__ float sh2[2];
    float lossn = 0.0f;
    const float invT = 1.0f / TNCE;

    // NCE A: q = HA-mean, neg = EB, att[:, :150]
    {
        float iq = invQA[n];
        float x0 = lposA[n] * invT;
        float mx = x0;
        for (int k = 0; k < K_; ++k) {
            float x = att[n * 300 + k] * dotA[n * K_ + k] * iq *
                      rsqrtf(nrmA[n * K_ + k]) * invT;
            mx = fmaxf(mx, x);
        }
        float se = expf(x0 - mx);
        for (int k = 0; k < K_; ++k) {
            float x = att[n * 300 + k] * dotA[n * K_ + k] * iq *
                      rsqrtf(nrmA[n * K_ + k]) * invT;
            se += expf(x - mx);
        }
        lossn += logf(se) + mx - x0;
    }
    // NCE B: q = HB-mean, neg = EA, att[:, 150:300]
    {
        float iq = invQB[n];
        float x0 = lposB[n] * invT;
        float mx = x0;
        for (int k = 0; k < K_; ++k) {
            float x = att[n * 300 + 150 + k] * dotB[n * K_ + k] * iq *
                      rsqrtf(nrmB[n * K_ + k]) * invT;
            mx = fmaxf(mx, x);
        }
        float se = expf(x0 - mx);
        for (int k = 0; k < K_; ++k) {
            float x = att[n * 300 + 150 + k] * dotB[n * K_ + k] * iq *
                      rsqrtf(nrmB[n * K_ + k]) * invT;
            se += expf(x - mx);
        }
        lossn += logf(se) + mx - x0;
    }
    float tot = reduce64(lossn, sh2);
    if (threadIdx.x == 0) {
        float snico = tot / (float)B_;
        float cls = acc[0];
        float rel = (acc[1] + acc[2]) / (float)(2 * B_ * TLEN_);
        float act = (acc[3] + (ETHETA / (float)B_) * acc[4]) * EALPHA;
        float total = cls + 0.01f * snico + act + THETA_ * rel;
        out[0] = total;
        out[1] = cls;
        out[2] = snico;
        out[3] = act;
        out[4] = rel;
    }
}

extern "C" void kernel_launch(void* const* d_in, const int* in_sizes, int n_in,
                              void* d_out, int out_size, void* d_ws, size_t ws_size,
                              hipStream_t stream) {
    (void)in_sizes; (void)n_in; (void)out_size; (void)ws_size;
    const float* video_scores = (const float*)d_in[0];
    const float* label        = (const float*)d_in[1];
    const float* HA           = (const float*)d_in[2];
    const float* EA           = (const float*)d_in[3];
    const float* HB           = (const float*)d_in[4];
    const float* EB           = (const float*)d_in[5];
    const float* re_s         = (const float*)d_in[6];
    const float* re_d         = (const float*)d_in[7];
    const float* actioness    = (const float*)d_in[8];
    const float* actioness_2  = (const float*)d_in[9];
    const float* att          = (const float*)d_in[10];
    float* out = (float*)d_out;

    float* ws   = (float*)d_ws;
    float* qA   = ws;                 // 64*2048
    float* kA   = qA + B_ * C_;
    float* qB   = kA + B_ * C_;
    float* kB   = qB + B_ * C_;
    float* dotA = kB + B_ * C_;       // 64*150
    float* nrmA = dotA + B_ * K_;
    float* dotB = nrmA + B_ * K_;
    float* nrmB = dotB + B_ * K_;
    float* lposA = nrmB + B_ * K_;    // 64
    float* lposB = lposA + B_;
    float* invQA = lposB + B_;
    float* invQB = invQA + B_;
    float* acc   = invQB + B_;        // 8 scalars

    k0_zero<<<1, 32, 0, stream>>>(acc);
    k1_means<<<dim3(C_ / 256, B_), 256, 0, stream>>>(HA, EA, HB, EB, qA, kA, qB, kB);
    k2_pos<<<B_, 256, 0, stream>>>(qA, kA, qB, kB, lposA, lposB, invQA, invQB);
    k3_gram<<<dim3(10, B_, 2), 32, 0, stream>>>(qA, qB, EA, EB, dotA, nrmA, dotB, nrmB);
    k4_act<<<TLEN_, 64, 0, stream>>>(actioness, actioness_2, acc);
    k5_cls<<<1, 256, 0, stream>>>(video_scores, label, acc);
    k6_rel<<<200, 256, 0, stream>>>(re_s, re_d, acc);
    k7_final<<<1, 64, 0, stream>>>(att, lposA, lposB, invQA, invQB,
                                   dotA, nrmA, dotB, nrmB, acc, out);
}